// CircleLossSun_79130477461848
// MI455X (gfx1250) — compile-verified
//
#include <hip/hip_runtime.h>
#include <hip/hip_bf16.h>
#include <math.h>

// Problem constants (fixed by the reference setup)
#define N_ROWS 4096
#define DIM    512
#define N_TILES (N_ROWS / 16)   // 256 16-row tiles
#define KCHUNKS (DIM / 4)       // 128 k-chunks of 4 for 16x16x4 f32 WMMA
#define KPAIRS  (KCHUNKS / 2)   // 64 pairs of k-chunks (one b128 per lane each)
#define NJ 8                    // consecutive j-tiles staged/consumed together

// TDM staging geometry
#define KGP 4                       // k-pairs per TDM stage
#define NKG (KPAIRS / KGP)          // 16 stages per j-group
#define JG_PER_BLOCK 2              // j-groups per block
#define STAGE_ROW_FLOATS (KGP * 128)        // 512 floats per j-tile per stage
#define STAGE_FLOATS (NJ * STAGE_ROW_FLOATS) // 4096 floats = 16 KB
#define STAGE_BYTES  (STAGE_FLOATS * 4)
#define JTILE_STRIDE_FLOATS (KPAIRS * 128)   // 8192 floats between j-tiles

typedef __attribute__((ext_vector_type(2))) float v2f;
typedef __attribute__((ext_vector_type(4))) float v4f;
typedef __attribute__((ext_vector_type(8))) float v8f;
typedef __attribute__((ext_vector_type(4))) unsigned int u32x4;
typedef __attribute__((ext_vector_type(8))) int i32x8;
typedef __attribute__((ext_vector_type(4))) int i32x4;

// ---------------------------------------------------------------------------
// Kernel 1: L2-normalize each row and store into a WMMA-fragment layout packed
// for b128 loads.  f32 16x16x4 A-fragment: lane l (l%16 = M row), k-half =
// l/16, holds feat[row][4c + 2*(l/16) + {0,1}].  Two chunks of a pair p are
// packed into 4 contiguous floats per lane:
//   featS[(((tile*64 + p)*32) + lane)*4 + (c&1)*2 + slot]
// so one b128 load (global or LDS) feeds two WMMAs, and a whole tile's
// fragment stream is contiguous (8192 floats) -> trivially TDM-describable.
// ---------------------------------------------------------------------------
__global__ __launch_bounds__(256) void normalize_swizzle_kernel(
    const float* __restrict__ in, float* __restrict__ featS,
    float* __restrict__ pos_sum, float* __restrict__ neg_sum)
{
    __shared__ float red[256];
    const int r = blockIdx.x;        // row 0..4095
    const int t = threadIdx.x;       // 0..255, 2 elements each

    float a0 = in[r * DIM + t];
    float a1 = in[r * DIM + t + 256];
    red[t] = a0 * a0 + a1 * a1;
    __syncthreads();
    for (int s = 128; s > 0; s >>= 1) {
        if (t < s) red[t] += red[t + s];
        __syncthreads();
    }
    const float inv = rsqrtf(red[0] + 1e-9f);   // matches /sqrt(ss + 1e-9)

    const int tile = r >> 4;
    const int rr   = r & 15;
    const int  ks[2] = { t, t + 256 };
    const float av[2] = { a0, a1 };
#pragma unroll
    for (int e = 0; e < 2; ++e) {
        const int k    = ks[e];
        const int c    = k >> 2;             // k-chunk 0..127
        const int kk   = k & 3;
        const int p    = c >> 1;             // chunk pair 0..63
        const int ce   = c & 1;              // which chunk within the pair
        const int lane = ((kk >> 1) << 4) | rr;  // 16*(k-half) + row-in-tile
        const int slot = (ce << 1) | (kk & 1);
        featS[((((tile * KPAIRS + p) * 32) + lane) << 2) + slot] = av[e] * inv;
    }
    if (t == 0) { pos_sum[r] = 0.0f; neg_sum[r] = 0.0f; }
}

// ---------------------------------------------------------------------------
// TDM descriptor issue: move one stage (8 j-tiles x KGP k-pairs) of B
// fragments, a 2D tile [8 rows x 512 floats, row stride 8192 floats], from
// global featS into LDS at byte offset ldsOff.  D# packed per CDNA5 ISA
// 8.3/8.4; groups 2/3 zero (tensor is 2D).  6-arg builtin form
// (amdgpu-toolchain / clang-23): trailing int32x8 group zeroed.
// ---------------------------------------------------------------------------
__device__ __forceinline__ void tdm_issue_stage(const float* featS, int jg,
                                                int kg, unsigned int ldsOff)
{
    const unsigned long long gaddr =
        (unsigned long long)(uintptr_t)featS +
        4ull * (unsigned long long)((jg * NJ) * JTILE_STRIDE_FLOATS +
                                    kg * STAGE_ROW_FLOATS);
    u32x4 g0;
    g0[0] = 1u;                                   // count=1, user descriptor
    g0[1] = ldsOff;                               // lds_addr (bytes)
    g0[2] = (unsigned int)gaddr;                  // global_addr[31:0]
    g0[3] = (unsigned int)((gaddr >> 32) & 0x01FFFFFFull) | (2u << 30); // type=2

    i32x8 g1;
    g1[0] = 0x00020000;                           // data_size=2 (4 bytes)
    g1[1] = (int)((unsigned)STAGE_ROW_FLOATS << 16);   // tensor_dim0 lo16
    g1[2] = (int)(8u << 16);                      // tensor_dim0 hi | tensor_dim1=8
    g1[3] = (int)((unsigned)STAGE_ROW_FLOATS << 16);   // tensor_dim1 hi | tile_dim0
    g1[4] = NJ;                                   // tile_dim1=8, tile_dim2=0
    g1[5] = JTILE_STRIDE_FLOATS;                  // tensor_dim0_stride lo32
    g1[6] = 0;                                    // stride hi, dim1_stride lo
    g1[7] = 0;

    i32x4 gz4 = {0, 0, 0, 0};
    i32x8 gz8 = {0, 0, 0, 0, 0, 0, 0, 0};
    __builtin_amdgcn_tensor_load_to_lds(g0, g1, gz4, gz4, gz8, 0);
}

// ---------------------------------------------------------------------------
// Kernel 2: fused sim = feat*feat^T (f32 WMMA 16x16x4) + circle-loss epilogue.
// Block = 8 waves; wave w owns i-tile blockIdx.x*8+w.  All waves share the
// same 8 consecutive j-tiles, whose B fragments are TDM-staged into a
// double-buffered LDS window (16 KB/stage) -> 8x on-chip reuse of B.
// Wave 0 drives the TDM pipeline (issue s+1, s_wait_tensorcnt, barrier).
// ---------------------------------------------------------------------------
__global__ __launch_bounds__(256) void circle_gemm_tdm_kernel(
    const float* __restrict__ featS, const long long* __restrict__ targets,
    float* __restrict__ pos_sum, float* __restrict__ neg_sum)
{
    extern __shared__ float smem[];   // 2 * STAGE_FLOATS (dynamic LDS, offset 0)

    const int tid   = threadIdx.x;
    const int lane  = tid & 31;
    const int wav   = tid >> 5;        // 0..7
    const int itile = blockIdx.x * 8 + wav;   // 0..255
    const int jg0   = blockIdx.y * JG_PER_BLOCK;
    const int half  = lane >> 4;       // C rows m = v + 8*half
    const int nlo   = lane & 15;       // C col within tile

    const v4f* __restrict__ fv = (const v4f*)featS;

    int tr[8];
#pragma unroll
    for (int v = 0; v < 8; ++v)
        tr[v] = (int)targets[itile * 16 + v + 8 * half];

    float epos[8], eneg[8];
#pragma unroll
    for (int v = 0; v < 8; ++v) { epos[v] = 0.0f; eneg[v] = 0.0f; }

    const int NSTAGE = JG_PER_BLOCK * NKG;
    if (wav == 0)
        tdm_issue_stage(featS, jg0, 0, 0u);

    int s = 0;
    for (int jgi = 0; jgi < JG_PER_BLOCK; ++jgi) {
        const int jg = jg0 + jgi;
        int tc[NJ];
#pragma unroll
        for (int jj = 0; jj < NJ; ++jj)
            tc[jj] = (int)targets[(jg * NJ + jj) * 16 + nlo];

        v8f acc[NJ];
#pragma unroll
        for (int jj = 0; jj < NJ; ++jj)
            acc[jj] = (v8f)0.0f;

        for (int kg = 0; kg < NKG; ++kg, ++s) {
            if (wav == 0) {
                const int ns = s + 1;
                if (ns < NSTAGE) {
                    tdm_issue_stage(featS, jg0 + ns / NKG, ns % NKG,
                                    (unsigned int)(ns & 1) * STAGE_BYTES);
                    __builtin_amdgcn_s_wait_tensorcnt(1);  // stage s landed
                } else {
                    __builtin_amdgcn_s_wait_tensorcnt(0);
                }
            }
            __syncthreads();           // stage s visible to all waves

            const float* buf = smem + (s & 1) * STAGE_FLOATS;
#pragma unroll
            for (int pr = 0; pr < KGP; ++pr) {
                const v4f a4 = fv[(itile * KPAIRS + kg * KGP + pr) * 32 + lane];
                v4f b4[NJ];
#pragma unroll
                for (int jj = 0; jj < NJ; ++jj)
                    b4[jj] = *(const v4f*)(buf + jj * STAGE_ROW_FLOATS +
                                           pr * 128 + lane * 4);

                const v2f alo = __builtin_shufflevector(a4, a4, 0, 1);
                const v2f ahi = __builtin_shufflevector(a4, a4, 2, 3);
#pragma unroll
                for (int jj = 0; jj < NJ; ++jj) {
                    const v2f blo = __builtin_shufflevector(b4[jj], b4[jj], 0, 1);
                    acc[jj] = __builtin_amdgcn_wmma_f32_16x16x4_f32(
                        false, alo, false, blo, (short)0, acc[jj], false, false);
                }
#pragma unroll
                for (int jj = 0; jj < NJ; ++jj) {
                    const v2f bhi = __builtin_shufflevector(b4[jj], b4[jj], 2, 3);
                    acc[jj] = __builtin_amdgcn_wmma_f32_16x16x4_f32(
                        false, ahi, false, bhi, (short)0, acc[jj], false, false);
                }
            }
            __syncthreads();           // all waves done reading buffer (s&1)
        }

        // Circle-loss elementwise transform for this j-group; accumulate into
        // per-row sums (row m = v + 8*half is independent of jj/jg).
        // m=0.35, gamma=96:
        //   pos: exp((s - 0.65) * 96*(1.35 - s))
        //   neg: exp((s - 0.35) * 96*max(s + 0.35, 0))
#pragma unroll
        for (int jj = 0; jj < NJ; ++jj) {
#pragma unroll
            for (int v = 0; v < 8; ++v) {
                const float sv = acc[jj][v];
                if (tr[v] == tc[jj])
                    epos[v] += __expf((sv - 0.65f) * (96.0f * (1.35f - sv)));
                else
                    eneg[v] += __expf((sv - 0.35f) * (96.0f * fmaxf(sv + 0.35f, 0.0f)));
            }
        }
    }

    // Reduce across the 16 lanes of each half (xor masks 1..8 stay in-half).
#pragma unroll
    for (int v = 0; v < 8; ++v) {
#pragma unroll
        for (int o = 8; o >= 1; o >>= 1) {
            epos[v] += __shfl_xor(epos[v], o, 32);
            eneg[v] += __shfl_xor(eneg[v], o, 32);
        }
    }
    if (nlo == 0) {
#pragma unroll
        for (int v = 0; v < 8; ++v) {
            const int m = itile * 16 + v + 8 * half;
            atomicAdd(&pos_sum[m], epos[v]);
            atomicAdd(&neg_sum[m], eneg[v]);
        }
    }
}

// ---------------------------------------------------------------------------
// Kernel 3: loss = mean(log1p(neg_sum / pos_sum))
// ---------------------------------------------------------------------------
__global__ __launch_bounds__(256) void finalize_kernel(
    const float* __restrict__ pos_sum, const float* __restrict__ neg_sum,
    float* __restrict__ out)
{
    __shared__ float red[256];
    const int t = threadIdx.x;
    float s = 0.0f;
    for (int r = t; r < N_ROWS; r += 256)
        s += log1pf(neg_sum[r] / pos_sum[r]);
    red[t] = s;
    __syncthreads();
    for (int w = 128; w > 0; w >>= 1) {
        if (t < w) red[t] += red[t + w];
        __syncthreads();
    }
    if (t == 0) out[0] = red[0] * (1.0f / (float)N_ROWS);
}

extern "C" void kernel_launch(void* const* d_in, const int* in_sizes, int n_in,
                              void* d_out, int out_size, void* d_ws, size_t ws_size,
                              hipStream_t stream)
{
    const float*     inputs  = (const float*)d_in[0];        // [4096, 512] f32
    const long long* targets = (const long long*)d_in[1];    // [4096] int64

    // Workspace layout: 8 MB swizzled feat + 2 x 16 KB row sums.
    float* featS   = (float*)d_ws;
    float* pos_sum = featS + (size_t)N_ROWS * DIM;
    float* neg_sum = pos_sum + N_ROWS;

    normalize_swizzle_kernel<<<N_ROWS, 256, 0, stream>>>(inputs, featS, pos_sum, neg_sum);

    // 32 i-groups (8 i-tiles each, one per wave) x 16 j-slices (2 j-groups each)
    dim3 g2(N_TILES / 8, 16, 1);
    circle_gemm_tdm_kernel<<<g2, 256, 2 * STAGE_BYTES, stream>>>(
        featS, targets, pos_sum, neg_sum);

    finalize_kernel<<<1, 256, 0, stream>>>(pos_sum, neg_sum, (float*)d_out);
}